// DendralNeuron_Dynamic_75926431858720
// MI455X (gfx1250) — compile-verified
//
#include <hip/hip_runtime.h>
#include <stdint.h>

// Tropical (min,+) contraction: out[b,d] = min_f min(x[b,f]-Wmin[d,f], Wmax[d,f]-x[b,f])
// B=1024, F=784, D=512. VALU-bound; TDM (tensor_load_to_lds) feeds LDS tiles,
// double-buffered, while all 8 waves run the sub/min3 stream.

typedef uint32_t u32;
typedef u32 u32x4 __attribute__((ext_vector_type(4)));
typedef int i32x4 __attribute__((ext_vector_type(4)));
typedef int i32x8 __attribute__((ext_vector_type(8)));

#define B_ 1024
#define F_ 784
#define D_ 512
#define KT 16            // k (f) elements per tile step
#define BT 64            // b rows per workgroup tile
#define DT 32            // d rows per workgroup tile
#define ROWSTRIDE 20     // floats per LDS row: 16 data + 4 pad (TDM pad_amount)
#define XTILE_FLOATS (BT * ROWSTRIDE)                  // 1280
#define WTILE_FLOATS (DT * ROWSTRIDE)                  // 640
#define BUF_FLOATS   (XTILE_FLOATS + 2 * WTILE_FLOATS) // 2560

// Issue one TDM 2-D tile load: tile_rows x KT f32 elements from a row-major
// [nrows_total x F_] matrix starting at (rowBase, kBase), into LDS at lds_byte.
// LDS padding: +4 DWORDs after every 16 DWORDs -> row stride 20 floats.
__device__ __forceinline__ void tdm_load_tile(u32 lds_byte, const float* base,
                                              u32 rowBase, u32 kBase,
                                              u32 nrows_total, u32 tile_rows) {
  uint64_t ga = (uint64_t)(uintptr_t)base + ((uint64_t)rowBase * F_ + kBase) * 4ull;

  u32x4 g0;
  g0[0] = 1u;                                   // count=1 (valid), user mode, no gather
  g0[1] = lds_byte;                             // lds_addr
  g0[2] = (u32)ga;                              // global_addr[31:0]
  g0[3] = ((u32)(ga >> 32) & 0x1FFFFFFu)        // global_addr[56:32]
          | 0x80000000u;                        // type=2 ("image")

  i32x8 g1;
  // data_size=2 (4B) | pad_enable | pad_interval=3 (16 DW) | pad_amount=3 (4 DW)
  g1[0] = (int)((2u << 16) | (1u << 20) | (3u << 22) | (3u << 25));
  g1[1] = (int)(((u32)F_ & 0xFFFFu) << 16);                 // tensor_dim0 lo16
  g1[2] = (int)((((u32)F_ >> 16) & 0xFFFFu)                 // tensor_dim0 hi16
                | ((nrows_total & 0xFFFFu) << 16));         // tensor_dim1 lo16
  g1[3] = (int)(((nrows_total >> 16) & 0xFFFFu)             // tensor_dim1 hi16
                | ((u32)KT << 16));                         // tile_dim0
  g1[4] = (int)tile_rows;                                   // tile_dim1 (tile_dim2=0)
  g1[5] = (int)F_;                                          // tensor_dim0_stride lo32
  g1[6] = 0;                                                // stride0 hi16, stride1 lo16
  g1[7] = 0;                                                // stride1 hi32

  i32x4 z4 = {0, 0, 0, 0};
#if defined(__clang_major__) && (__clang_major__ >= 23)
  i32x8 z8 = {0, 0, 0, 0, 0, 0, 0, 0};
  __builtin_amdgcn_tensor_load_to_lds(g0, g1, z4, z4, z8, 0);
#else
  __builtin_amdgcn_tensor_load_to_lds(g0, g1, z4, z4, 0);
#endif
}

__global__ __launch_bounds__(256, 2)
void dendral_minplus_kernel(const float* __restrict__ x,
                            const float* __restrict__ wmin,
                            const float* __restrict__ wmax,
                            float* __restrict__ out) {
  __shared__ __align__(16) float smem[2 * BUF_FLOATS];

  const u32 t  = threadIdx.x;
  const u32 tx = t & 15u;   // d lane: covers d = tx, tx+16
  const u32 ty = t >> 4;    // b lane: covers b = ty + 16*i, i=0..3
  const u32 dBase = blockIdx.x * DT;
  const u32 bBase = blockIdx.y * BT;
  const u32 ldsBase = (u32)(uintptr_t)(void*)smem;  // flat LDS addr low 32 = LDS offset
  const bool issuer = (t < 32);                     // wave 0 drives the TDM

  float acc[4][2];
#pragma unroll
  for (int i = 0; i < 4; ++i)
#pragma unroll
    for (int j = 0; j < 2; ++j) acc[i][j] = __builtin_inff();

  const int NKT = F_ / KT;  // 49, exact

  // Prologue: fill buffer 0
  if (issuer) {
    tdm_load_tile(ldsBase, x, bBase, 0, B_, BT);
    tdm_load_tile(ldsBase + XTILE_FLOATS * 4u, wmin, dBase, 0, D_, DT);
    tdm_load_tile(ldsBase + (XTILE_FLOATS + WTILE_FLOATS) * 4u, wmax, dBase, 0, D_, DT);
    __builtin_amdgcn_s_wait_tensorcnt(0);
  }
  __syncthreads();

  for (int kt = 0; kt < NKT; ++kt) {
    const u32 cur = (u32)(kt & 1);
    const u32 nxt = cur ^ 1u;

    // Kick off next tile's DMA into the other buffer (its previous consumers
    // all passed the barrier at the end of iteration kt-1).
    if (issuer && (kt + 1 < NKT)) {
      const u32 off = ldsBase + nxt * BUF_FLOATS * 4u;
      const u32 kB = (u32)(kt + 1) * KT;
      tdm_load_tile(off, x, bBase, kB, B_, BT);
      tdm_load_tile(off + XTILE_FLOATS * 4u, wmin, dBase, kB, D_, DT);
      tdm_load_tile(off + (XTILE_FLOATS + WTILE_FLOATS) * 4u, wmax, dBase, kB, D_, DT);
    }

    const float* Xs = smem + cur * BUF_FLOATS;
    const float* Ns = Xs + XTILE_FLOATS;
    const float* Ms = Ns + WTILE_FLOATS;

#pragma unroll
    for (int kk = 0; kk < KT; kk += 4) {
      float4 a[4], n[2], m[2];
#pragma unroll
      for (int i = 0; i < 4; ++i)
        a[i] = *(const float4*)&Xs[(ty + 16u * i) * ROWSTRIDE + kk];
#pragma unroll
      for (int j = 0; j < 2; ++j) {
        n[j] = *(const float4*)&Ns[(tx + 16u * j) * ROWSTRIDE + kk];
        m[j] = *(const float4*)&Ms[(tx + 16u * j) * ROWSTRIDE + kk];
      }
#pragma unroll
      for (int i = 0; i < 4; ++i) {
#pragma unroll
        for (int j = 0; j < 2; ++j) {
          // min-accumulate both dendrites; 2x v_sub + v_min3 per f element
          acc[i][j] = fminf(fminf(a[i].x - n[j].x, m[j].x - a[i].x), acc[i][j]);
          acc[i][j] = fminf(fminf(a[i].y - n[j].y, m[j].y - a[i].y), acc[i][j]);
          acc[i][j] = fminf(fminf(a[i].z - n[j].z, m[j].z - a[i].z), acc[i][j]);
          acc[i][j] = fminf(fminf(a[i].w - n[j].w, m[j].w - a[i].w), acc[i][j]);
        }
      }
    }

    if (issuer) __builtin_amdgcn_s_wait_tensorcnt(0);
    __syncthreads();
  }

  // Write 4x2 register tile: lanes tx=0..15 give 16 consecutive d per store group
#pragma unroll
  for (int i = 0; i < 4; ++i) {
    const u32 b = bBase + ty + 16u * i;
#pragma unroll
    for (int j = 0; j < 2; ++j) {
      const u32 d = dBase + tx + 16u * j;
      out[(size_t)b * D_ + d] = acc[i][j];
    }
  }
}

extern "C" void kernel_launch(void* const* d_in, const int* in_sizes, int n_in,
                              void* d_out, int out_size, void* d_ws, size_t ws_size,
                              hipStream_t stream) {
  (void)in_sizes; (void)n_in; (void)d_ws; (void)ws_size; (void)out_size;
  const float* x    = (const float*)d_in[0];
  const float* wmin = (const float*)d_in[1];
  const float* wmax = (const float*)d_in[2];
  float* out = (float*)d_out;

  dim3 grid(D_ / DT, B_ / BT);  // (16, 16) = 256 workgroups
  dendral_minplus_kernel<<<grid, 256, 0, stream>>>(x, wmin, wmax, out);
}